// ModelClass_22711787061488
// MI455X (gfx1250) — compile-verified
//
#include <hip/hip_runtime.h>

typedef __attribute__((ext_vector_type(2))) float v2f;
typedef __attribute__((ext_vector_type(8))) float v8f;

#define G    256
#define P    512
#define F    4
#define KNN  6
#define NTOT (G * P)

__device__ __forceinline__ float prelu(float v, float a) {
    return v >= 0.0f ? v : a * v;
}

// ---------------------------------------------------------------------------
// kNN per graph: Gram matrix via V_WMMA_F32_16X16X4_F32, top-6 per row.
// One workgroup (8 wave32) per graph; each wave owns 16-row strips.
// Symmetry trick: compute tile(cb, rb) = A(col-block) x B(my row-block) so the
// C fragment lands row-major per lane: lane (lm, half) holds
// gram[myrow][cb*16 + 8*half + v] in c[v] -- no LDS transpose needed.
// ---------------------------------------------------------------------------
__global__ __launch_bounds__(256)
void knn_kernel(const float* __restrict__ x, int* __restrict__ srcIdx)
{
    __shared__ __align__(16) float Xs[P * F];   // 8 KB: graph nodes [p][f]
    __shared__ __align__(16) float sqn[P];      // 2 KB: squared norms
    __shared__ float md[8][32][KNN];            // merge buffers
    __shared__ int   mi[8][32][KNN];

    const int g   = blockIdx.x;
    const int tid = threadIdx.x;
    const float* xg = x + (size_t)g * P * F;

    for (int i = tid; i < P * F; i += 256) Xs[i] = xg[i];
    __syncthreads();
    for (int p = tid; p < P; p += 256) {
        float s = 0.0f;
#pragma unroll
        for (int f = 0; f < F; ++f) { float v = Xs[p * F + f]; s += v * v; }
        sqn[p] = s;
    }
    __syncthreads();

    const int wave = tid >> 5;
    const int lane = tid & 31;
    const int half = lane >> 4;        // 0: K={0,1}, cols v+0 ; 1: K={2,3}, cols v+8
    const int lm   = lane & 15;
    const int k0   = half * 2;

    for (int it = 0; it < 4; ++it) {
        const int rb      = wave * 4 + it;   // row-block 0..31
        const int rowbase = rb * 16;

        // B fragment = my row block (fixed across all column tiles)
        v2f b;
        b[0] = Xs[(rowbase + lm) * F + k0];
        b[1] = Xs[(rowbase + lm) * F + k0 + 1];

        float bd[KNN];
        int   bi[KNN];
#pragma unroll
        for (int j = 0; j < KNN; ++j) { bd[j] = 3.0e38f; bi[j] = 0; }

        const int   myrow = rowbase + lm;    // row this lane selects for
        const float sqr   = sqn[myrow];

        for (int cb = 0; cb < 32; ++cb) {
            // A fragment = column block cb
            v2f a;
            a[0] = Xs[(cb * 16 + lm) * F + k0];
            a[1] = Xs[(cb * 16 + lm) * F + k0 + 1];

            v8f c = {};
            c = __builtin_amdgcn_wmma_f32_16x16x4_f32(
                    false, a, false, b, (short)0, c, false, false);

            // lane's 8 contiguous columns this tile
            const int colbase = cb * 16 + 8 * half;
            const float4 sq0 = *(const float4*)&sqn[colbase];
            const float4 sq1 = *(const float4*)&sqn[colbase + 4];
            float sqv[8] = { sq0.x, sq0.y, sq0.z, sq0.w,
                             sq1.x, sq1.y, sq1.z, sq1.w };

#pragma unroll
            for (int v = 0; v < 8; ++v) {
                const int qcol = colbase + v;
                float d = sqr + sqv[v] - 2.0f * c[v];
                if (qcol == myrow) d = 3.3e38f;   // exclude self
                if (d < bd[KNN - 1]) {
                    bd[KNN - 1] = d; bi[KNN - 1] = qcol;
#pragma unroll
                    for (int j = KNN - 1; j > 0; --j) {
                        if (bd[j] < bd[j - 1]) {
                            float td = bd[j]; bd[j] = bd[j - 1]; bd[j - 1] = td;
                            int   ti = bi[j]; bi[j] = bi[j - 1]; bi[j - 1] = ti;
                        }
                    }
                }
            }
        }

        // merge the two half-lane candidate lists (same-wave LDS is in-order)
#pragma unroll
        for (int j = 0; j < KNN; ++j) { md[wave][lane][j] = bd[j]; mi[wave][lane][j] = bi[j]; }
        if (half == 0) {
#pragma unroll
            for (int j = 0; j < KNN; ++j) {
                const float dc = md[wave][lm + 16][j];
                const int   ic = mi[wave][lm + 16][j];
                if (dc < bd[KNN - 1]) {
                    bd[KNN - 1] = dc; bi[KNN - 1] = ic;
#pragma unroll
                    for (int jj = KNN - 1; jj > 0; --jj) {
                        if (bd[jj] < bd[jj - 1]) {
                            float td = bd[jj]; bd[jj] = bd[jj - 1]; bd[jj - 1] = td;
                            int   ti = bi[jj]; bi[jj] = bi[jj - 1]; bi[jj - 1] = ti;
                        }
                    }
                }
            }
            const int nrow = g * P + myrow;
#pragma unroll
            for (int j = 0; j < KNN; ++j)
                srcIdx[nrow * KNN + j] = g * P + bi[j];
        }
    }
}

// ---------------------------------------------------------------------------
// pre_nn: Linear-PReLU-Linear-PReLU, plus per-block partial sums for BN stats.
// ---------------------------------------------------------------------------
__global__ __launch_bounds__(256)
void prenn_kernel(const float* __restrict__ x,
                  const float* __restrict__ W1, const float* __restrict__ b1,
                  const float* __restrict__ a1,
                  const float* __restrict__ W2, const float* __restrict__ b2,
                  const float* __restrict__ a2,
                  float* __restrict__ hpre, float* __restrict__ partials)
{
    __shared__ float red[256 * 8];
    const int tid = threadIdx.x;
    const int n   = blockIdx.x * 256 + tid;

    const float4 xv4 = ((const float4*)x)[n];
    const float xv[F] = { xv4.x, xv4.y, xv4.z, xv4.w };
    float h1[F], h2[F];
#pragma unroll
    for (int o = 0; o < F; ++o) {
        float t = b1[o];
#pragma unroll
        for (int f = 0; f < F; ++f) t += xv[f] * W1[f * F + o];
        h1[o] = prelu(t, a1[o]);
    }
#pragma unroll
    for (int o = 0; o < F; ++o) {
        float t = b2[o];
#pragma unroll
        for (int f = 0; f < F; ++f) t += h1[f] * W2[f * F + o];
        h2[o] = prelu(t, a2[o]);
    }
    float4 hv4; hv4.x = h2[0]; hv4.y = h2[1]; hv4.z = h2[2]; hv4.w = h2[3];
    ((float4*)hpre)[n] = hv4;
#pragma unroll
    for (int o = 0; o < F; ++o) {
        red[tid * 8 + o]     = h2[o];
        red[tid * 8 + 4 + o] = h2[o] * h2[o];
    }
    __syncthreads();
    for (int off = 128; off > 0; off >>= 1) {
        if (tid < off) {
#pragma unroll
            for (int k = 0; k < 8; ++k)
                red[tid * 8 + k] += red[(tid + off) * 8 + k];
        }
        __syncthreads();
    }
    if (tid < 8) partials[blockIdx.x * 8 + tid] = red[tid];
}

// ---------------------------------------------------------------------------
// Reduce 512 block-partials -> mu[4], rstd[4].
// ---------------------------------------------------------------------------
__global__ __launch_bounds__(256)
void stats_kernel(const float* __restrict__ partials, float* __restrict__ stats)
{
    __shared__ float red[256 * 8];
    const int tid = threadIdx.x;
#pragma unroll
    for (int k = 0; k < 8; ++k)
        red[tid * 8 + k] = partials[tid * 8 + k] + partials[(tid + 256) * 8 + k];
    __syncthreads();
    for (int off = 128; off > 0; off >>= 1) {
        if (tid < off) {
#pragma unroll
            for (int k = 0; k < 8; ++k)
                red[tid * 8 + k] += red[(tid + off) * 8 + k];
        }
        __syncthreads();
    }
    if (tid == 0) {
        const float invN = 1.0f / (float)NTOT;
#pragma unroll
        for (int f = 0; f < F; ++f) {
            const float mu  = red[f] * invN;
            const float ex2 = red[4 + f] * invN;
            const float var = ex2 - mu * mu;
            stats[f]     = mu;
            stats[4 + f] = rsqrtf(var + 1e-5f);
        }
    }
}

__global__ __launch_bounds__(256)
void bnapply_kernel(const float* __restrict__ hpre, const float* __restrict__ stats,
                    const float* __restrict__ gg, const float* __restrict__ bb,
                    float* __restrict__ H0)
{
    const int n = blockIdx.x * 256 + threadIdx.x;
    const float4 hv = ((const float4*)hpre)[n];
    const float h[F] = { hv.x, hv.y, hv.z, hv.w };
    float4 o4;
    float o[F];
#pragma unroll
    for (int f = 0; f < F; ++f)
        o[f] = (h[f] - stats[f]) * stats[4 + f] * gg[f] + bb[f];
    o4.x = o[0]; o4.y = o[1]; o4.z = o[2]; o4.w = o[3];
    ((float4*)H0)[n] = o4;
}

// ---------------------------------------------------------------------------
// GeneralConv layer: h' = prelu( (sum_j h[src_j]) @ W + 6*b + h, act_a )
// ---------------------------------------------------------------------------
__global__ __launch_bounds__(256)
void conv_kernel(const float* __restrict__ Hin, float* __restrict__ Hout,
                 const int* __restrict__ srcIdx,
                 const float* __restrict__ Wm, const float* __restrict__ bm,
                 const float* __restrict__ act_a)
{
    const int n = blockIdx.x * 256 + threadIdx.x;
    const float4* Hin4 = (const float4*)Hin;

    float nb[F] = {0.0f, 0.0f, 0.0f, 0.0f};
#pragma unroll
    for (int j = 0; j < KNN; ++j) {
        const int s = srcIdx[n * KNN + j];
        const float4 hv = Hin4[s];                 // one B128 gather
        nb[0] += hv.x; nb[1] += hv.y; nb[2] += hv.z; nb[3] += hv.w;
    }
    const float4 hs4 = Hin4[n];
    const float hs[F] = { hs4.x, hs4.y, hs4.z, hs4.w };
    float o[F];
#pragma unroll
    for (int o_ = 0; o_ < F; ++o_) {
        float acc = 6.0f * bm[o_];
#pragma unroll
        for (int f = 0; f < F; ++f) acc += nb[f] * Wm[f * F + o_];
        o[o_] = prelu(acc + hs[o_], act_a[o_]);
    }
    float4 o4; o4.x = o[0]; o4.y = o[1]; o4.z = o[2]; o4.w = o[3];
    ((float4*)Hout)[n] = o4;
}

// ---------------------------------------------------------------------------
// JK-cat + global_add_pool + 2-layer head. One workgroup per graph.
// ---------------------------------------------------------------------------
__global__ __launch_bounds__(256)
void head_kernel(const float* __restrict__ H,
                 const float* __restrict__ W1, const float* __restrict__ b1,
                 const float* __restrict__ W2, const float* __restrict__ b2,
                 float* __restrict__ out)
{
    __shared__ float red[256 * 20];
    __shared__ float t1[20];
    const int g   = blockIdx.x;
    const int tid = threadIdx.x;
    const float4* H4 = (const float4*)H;   // [layer][node] of float4

    float acc[20];
#pragma unroll
    for (int k = 0; k < 20; ++k) acc[k] = 0.0f;
    for (int p = tid; p < P; p += 256) {
        const int n = g * P + p;
#pragma unroll
        for (int l = 0; l < 5; ++l) {
            const float4 hv = H4[(size_t)l * NTOT + n];
            acc[l * F + 0] += hv.x;
            acc[l * F + 1] += hv.y;
            acc[l * F + 2] += hv.z;
            acc[l * F + 3] += hv.w;
        }
    }
#pragma unroll
    for (int k = 0; k < 20; ++k) red[tid * 20 + k] = acc[k];
    __syncthreads();
    for (int off = 128; off > 0; off >>= 1) {
        if (tid < off) {
#pragma unroll
            for (int k = 0; k < 20; ++k)
                red[tid * 20 + k] += red[(tid + off) * 20 + k];
        }
        __syncthreads();
    }
    if (tid < 20) {
        float a = b1[tid];
#pragma unroll
        for (int i = 0; i < 20; ++i) a += red[i] * W1[i * 20 + tid];
        t1[tid] = a > 0.0f ? a : 0.0f;
    }
    __syncthreads();
    if (tid == 0) {
        float o = b2[0];
#pragma unroll
        for (int j = 0; j < 20; ++j) o += t1[j] * W2[j];
        out[g] = o;
    }
}

// ---------------------------------------------------------------------------
extern "C" void kernel_launch(void* const* d_in, const int* in_sizes, int n_in,
                              void* d_out, int out_size, void* d_ws, size_t ws_size,
                              hipStream_t stream)
{
    const float* x     = (const float*)d_in[0];
    /* d_in[1] = batch (int32), implicit n/P, unused */
    const float* preW1 = (const float*)d_in[2];
    const float* preb1 = (const float*)d_in[3];
    const float* prea1 = (const float*)d_in[4];
    const float* preW2 = (const float*)d_in[5];
    const float* preb2 = (const float*)d_in[6];
    const float* prea2 = (const float*)d_in[7];
    const float* bng   = (const float*)d_in[8];
    const float* bnb   = (const float*)d_in[9];
    const float* acta  = (const float*)d_in[10];
    const float* convW = (const float*)d_in[11];   // [4][F][F]
    const float* convb = (const float*)d_in[12];   // [4][F]
    const float* hW1   = (const float*)d_in[13];   // [20][20]
    const float* hb1   = (const float*)d_in[14];
    const float* hW2   = (const float*)d_in[15];   // [20][1]
    const float* hb2   = (const float*)d_in[16];

    float* ws       = (float*)d_ws;
    float* H        = ws;                               // 5 * NTOT * F floats
    float* hpre     = H + (size_t)NTOT * F;             // reuse layer-1 slot
    float* partials = H + (size_t)5 * NTOT * F;         // 512 * 8
    float* stats    = partials + 512 * 8;               // 8
    int*   srcIdx   = (int*)(stats + 8);                // NTOT * KNN

    knn_kernel  <<<G,        256, 0, stream>>>(x, srcIdx);
    prenn_kernel<<<NTOT/256, 256, 0, stream>>>(x, preW1, preb1, prea1,
                                               preW2, preb2, prea2, hpre, partials);
    stats_kernel<<<1,        256, 0, stream>>>(partials, stats);
    bnapply_kernel<<<NTOT/256, 256, 0, stream>>>(hpre, stats, bng, bnb, H);
    for (int i = 0; i < 4; ++i)
        conv_kernel<<<NTOT/256, 256, 0, stream>>>(H + (size_t)i * NTOT * F,
                                                  H + (size_t)(i + 1) * NTOT * F,
                                                  srcIdx, convW + i * F * F,
                                                  convb + i * F, acta);
    head_kernel <<<G,        256, 0, stream>>>(H, hW1, hb1, hW2, hb2, (float*)d_out);
}